// Attention_70514773066385
// MI455X (gfx1250) — compile-verified
//
#include <hip/hip_runtime.h>
#include <hip/hip_bf16.h>
#include <math.h>

#define HEADS 12
#define HD    64
#define HIDN  768
#define SEQ   2048
#define CHUNK 512   // key chunk held in LDS: 16 x 512 f32 = 32 KB

typedef float v2f __attribute__((ext_vector_type(2)));
typedef float v8f __attribute__((ext_vector_type(8)));

static __device__ __forceinline__ v8f zero8() {
  v8f z = {0.f, 0.f, 0.f, 0.f, 0.f, 0.f, 0.f, 0.f};
  return z;
}

// D = A(16x4, f32) * B(4x16, f32) + C(16x16, f32)  ->  v_wmma_f32_16x16x4_f32
static __device__ __forceinline__ v8f wmma4(v2f a, v2f b, v8f c) {
  return __builtin_amdgcn_wmma_f32_16x16x4_f32(false, a, false, b, (short)0, c,
                                               false, false);
}

// ---------------------------------------------------------------------------
// C(2048x768) = A(2048x768) @ B(768x768), all row-major f32.
// 1 wave computes a 16x64 strip (4 WMMA C tiles); A-frag shared across tiles.
// grid: 192 blocks x 256 threads = 1536 waves = (2048/16)*(768/64) tasks.
// ---------------------------------------------------------------------------
__global__ __launch_bounds__(256) void gemm_wmma_f32(
    const float* __restrict__ A, const float* __restrict__ B,
    float* __restrict__ C) {
  const int lane  = threadIdx.x & 31;
  const int wave  = blockIdx.x * 8 + (threadIdx.x >> 5);
  const int mbase = (wave / (HIDN / 64)) * 16;
  const int nbase = (wave % (HIDN / 64)) * 64;
  const int idx   = lane & 15;
  const int hi    = lane >> 4;
  const int kk    = hi << 1;

  v8f c0 = zero8(), c1 = zero8(), c2 = zero8(), c3 = zero8();
  const float* arow = A + (size_t)(mbase + idx) * HIDN + kk;
  const float* bcol = B + nbase + idx;

  for (int kb = 0; kb < HIDN; kb += 4) {
    v2f a = *(const v2f*)(arow + kb);                 // A[m][kb+kk .. +1]
    const float* bp = bcol + (size_t)(kb + kk) * HIDN;
    v2f b0 = {bp[0],  bp[HIDN]};                      // B[kb+kk..+1][n..]
    v2f b1 = {bp[16], bp[16 + HIDN]};
    v2f b2 = {bp[32], bp[32 + HIDN]};
    v2f b3 = {bp[48], bp[48 + HIDN]};
    c0 = wmma4(a, b0, c0);
    c1 = wmma4(a, b1, c1);
    c2 = wmma4(a, b2, c2);
    c3 = wmma4(a, b3, c3);
  }
#pragma unroll
  for (int r = 0; r < 8; ++r) {
    float* crow = C + (size_t)(mbase + r + 8 * hi) * HIDN + nbase + idx;
    crow[0]  = c0[r];
    crow[16] = c1[r];
    crow[32] = c2[r];
    crow[48] = c3[r];
  }
}

// ---------------------------------------------------------------------------
// Attention: 1 block (8 waves) = one (head, 16-query block).
// Online (flash-style) softmax over 4 key chunks of 512; score tile in LDS.
// Writes raw (pre-mask scores incl. gated norm(s_prev)) with NT stores, and
// ctx = softmax(raw+mask) @ V into workspace.
// ---------------------------------------------------------------------------
__global__ __launch_bounds__(256) void attn_wmma(
    const float* __restrict__ Q, const float* __restrict__ K,
    const float* __restrict__ V, const float* __restrict__ s_prev,
    const float* __restrict__ alpha, const float* __restrict__ gamma,
    float* __restrict__ raw, float* __restrict__ ctx) {
  __shared__ float Sb[16 * CHUNK];  // score/w tile; reused for ctx partials
  __shared__ float muS[16], rsS[16], Mrow[16], denRow[16], facRow[16];

  const int qb    = blockIdx.x;
  const int h     = blockIdx.y;
  const int qbase = qb * 16;
  const int t     = threadIdx.x;
  const int lane  = t & 31;
  const int wv    = t >> 5;  // 0..7
  const int idx   = lane & 15;
  const int hi    = lane >> 4;
  const int kk    = hi << 1;
  const float NEG_INF = -__builtin_inff();

  const float g = gamma[h] / (1.0f + __expf(-alpha[h]));  // sigmoid*gamma
  const size_t srow_base = ((size_t)h * SEQ + qbase) * SEQ;

  // --- per-row mean / rstd of s_prev (full 2048 row), 16 lanes per row ---
  {
    const int r = t >> 4, ln = t & 15;
    const float* sp = s_prev + srow_base + (size_t)r * SEQ;
    float sum = 0.f, sq = 0.f;
#pragma unroll 4
    for (int i = 0; i < SEQ / 16; ++i) {
      float v = sp[ln + 16 * i];
      sum += v;
      sq = fmaf(v, v, sq);
    }
    for (int m = 8; m >= 1; m >>= 1) {
      sum += __shfl_xor(sum, m);
      sq  += __shfl_xor(sq, m);
    }
    if (ln == 0) {
      float mu  = sum * (1.0f / SEQ);
      float var = sq * (1.0f / SEQ) - mu * mu;
      muS[r]    = mu;
      rsS[r]    = rsqrtf(var + 1e-6f);
      Mrow[r]   = NEG_INF;
      denRow[r] = 0.f;
    }
  }
  __syncthreads();

  // --- preload Q A-fragments for all 16 k-steps of D=64 ---
  v2f qa[16];
  {
    const float* qrow = Q + (size_t)(qbase + idx) * HIDN + h * HD + kk;
#pragma unroll
    for (int s = 0; s < 16; ++s) qa[s] = *(const v2f*)(qrow + 4 * s);
  }

  v8f c0 = zero8(), c1 = zero8(), c2 = zero8(), c3 = zero8();  // ctx accum

  for (int ch = 0; ch < SEQ / CHUNK; ++ch) {
    const int k0 = ch * CHUNK;

    // --- scores: each wave computes 4 key tiles of 16 (QK^T / 8) ---
#pragma unroll
    for (int j = 0; j < CHUNK / (16 * 8); ++j) {
      const int jt = wv * (CHUNK / (16 * 8)) + j;
      const float* krow =
          K + (size_t)(k0 + jt * 16 + idx) * HIDN + h * HD + kk;
      v8f acc = zero8();
#pragma unroll
      for (int s = 0; s < 16; ++s) {
        v2f kb = *(const v2f*)(krow + 4 * s);
        acc = wmma4(qa[s], kb, acc);
      }
#pragma unroll
      for (int r = 0; r < 8; ++r)
        Sb[(r + 8 * hi) * CHUNK + jt * 16 + idx] = acc[r] * 0.125f;
    }
    __syncthreads();

    // --- fused: + g*norm(s_prev), NT-store raw, online softmax weights ---
    {
      const int r = t >> 4, ln = t & 15;
      const int qg = qbase + r;
      const float mu = muS[r], rs = rsS[r];
      const float* sp = s_prev + srow_base + (size_t)r * SEQ + k0;
      float* rawrow   = raw    + srow_base + (size_t)r * SEQ + k0;
      float mx = NEG_INF;
#pragma unroll 4
      for (int i = 0; i < CHUNK / 16; ++i) {
        const int kl = ln + 16 * i;
        float v = Sb[r * CHUNK + kl] + g * (sp[kl] - mu) * rs;
        __builtin_nontemporal_store(v, rawrow + kl);  // raw is streamed out
        Sb[r * CHUNK + kl] = v;
        if (k0 + kl <= qg) mx = fmaxf(mx, v);         // causal-masked max
      }
      for (int m = 8; m >= 1; m >>= 1) mx = fmaxf(mx, __shfl_xor(mx, m));
      if (ln == 0) {
        float newM = fmaxf(Mrow[r], mx);              // finite after chunk 0
        float f    = __expf(Mrow[r] - newM);
        facRow[r]  = f;
        denRow[r] *= f;
        Mrow[r]    = newM;
      }
      const float newM = Mrow[r];  // same-wave LDS broadcast (in-order DS)
      float psum = 0.f;
#pragma unroll 4
      for (int i = 0; i < CHUNK / 16; ++i) {
        const int kl = ln + 16 * i;
        float w = (k0 + kl <= qg) ? __expf(Sb[r * CHUNK + kl] - newM) : 0.f;
        Sb[r * CHUNK + kl] = w;
        psum += w;
      }
      for (int m = 8; m >= 1; m >>= 1) psum += __shfl_xor(psum, m);
      if (ln == 0) denRow[r] += psum;
    }
    __syncthreads();

    // --- rescale ctx accumulators, then ctx += w_chunk @ V_chunk ---
#pragma unroll
    for (int r = 0; r < 8; ++r) {
      const float f = facRow[r + 8 * hi];
      c0[r] *= f; c1[r] *= f; c2[r] *= f; c3[r] *= f;
    }
#pragma unroll
    for (int s = 0; s < 16; ++s) {
      const int kb = wv * 64 + 4 * s;  // this wave's 64-wide k slice
      v2f a = *(const v2f*)(&Sb[idx * CHUNK + kb + kk]);  // w[m][k..k+1]
      const float* v0 = V + (size_t)(k0 + kb + kk) * HIDN + h * HD + idx;
      v2f b0 = {v0[0],  v0[HIDN]};
      v2f b1 = {v0[16], v0[16 + HIDN]};
      v2f b2 = {v0[32], v0[32 + HIDN]};
      v2f b3 = {v0[48], v0[48 + HIDN]};
      c0 = wmma4(a, b0, c0);
      c1 = wmma4(a, b1, c1);
      c2 = wmma4(a, b2, c2);
      c3 = wmma4(a, b3, c3);
    }
    __syncthreads();  // before next chunk overwrites Sb
  }

  // --- cross-wave reduction of ctx partials (reuse Sb: 8*16*64 = 8192 f32) ---
#pragma unroll
  for (int r = 0; r < 8; ++r) {
    float* p = &Sb[wv * 1024 + (r + 8 * hi) * 64 + idx];
    p[0]  = c0[r];
    p[16] = c1[r];
    p[32] = c2[r];
    p[48] = c3[r];
  }
  __syncthreads();
#pragma unroll
  for (int i = 0; i < 4; ++i) {
    const int o = t + 256 * i;  // 0..1023 over 16x64 outputs
    const int m = o >> 6, d = o & 63;
    float acc = 0.f;
#pragma unroll
    for (int w = 0; w < 8; ++w) acc += Sb[w * 1024 + o];
    ctx[(size_t)(qbase + m) * HIDN + h * HD + d] = acc / denRow[m];
  }
}

// ---------------------------------------------------------------------------
extern "C" void kernel_launch(void* const* d_in, const int* in_sizes, int n_in,
                              void* d_out, int out_size, void* d_ws,
                              size_t ws_size, hipStream_t stream) {
  const float* x      = (const float*)d_in[0];
  const float* s_prev = (const float*)d_in[1];
  const float* Wq     = (const float*)d_in[2];
  const float* Wk     = (const float*)d_in[3];
  const float* Wv     = (const float*)d_in[4];
  const float* Wo     = (const float*)d_in[5];
  const float* alpha  = (const float*)d_in[6];
  const float* gamma  = (const float*)d_in[7];

  float* y   = (float*)d_out;                 // (1, 2048, 768)
  float* raw = y + (size_t)SEQ * HIDN;        // (1, 12, 2048, 2048)

  float* ws = (float*)d_ws;
  float* Qp = ws;                             // (2048, 768) each
  float* Kp = Qp + (size_t)SEQ * HIDN;
  float* Vp = Kp + (size_t)SEQ * HIDN;
  float* Cp = Vp + (size_t)SEQ * HIDN;        // ctx (2048, 768)

  gemm_wmma_f32<<<192, 256, 0, stream>>>(x, Wq, Qp);
  gemm_wmma_f32<<<192, 256, 0, stream>>>(x, Wk, Kp);
  gemm_wmma_f32<<<192, 256, 0, stream>>>(x, Wv, Vp);
  attn_wmma<<<dim3(SEQ / 16, HEADS), 256, 0, stream>>>(
      Qp, Kp, Vp, s_prev, alpha, gamma, raw, Cp);
  gemm_wmma_f32<<<192, 256, 0, stream>>>(Cp, Wo, y);
}